// YOLOLoss_12120397709546
// MI455X (gfx1250) — compile-verified
//
#include <hip/hip_runtime.h>
#include <hip/hip_bf16.h>
#include <math.h>

// Problem dims fixed by the reference setup_inputs():
#define BATCH 64
#define NA 3
#define NC 11
#define NH 80
#define NW 80
#define NT 50
#define NCH (5 + NC)              // 16 channels per anchor
#define CHTOT (NA * NCH)          // 48
#define PLANE (NH * NW)           // 6400
#define NPLANES (BATCH * NA)      // 192 obj planes
#define NTOT (BATCH * NA * NH * NW)
#define NTGT (BATCH * NT)         // 3200 targets
#define EPSF 1e-7f
#define INV_PI2_4 0.4052847345693511f  // 4/pi^2

typedef float v2f __attribute__((ext_vector_type(2)));
typedef float v8f __attribute__((ext_vector_type(8)));

__device__ __forceinline__ float softplusf(float x) {
    // logaddexp(0, x) = max(x,0) + log1p(exp(-|x|)), numerically stable
    return fmaxf(x, 0.0f) + log1pf(expf(-fabsf(x)));
}
__device__ __forceinline__ float sigmoidf_(float x) {
    return 1.0f / (1.0f + expf(-x));
}

// ---------------------------------------------------------------------------
// Kernel 0: zero the objectness mask bit-array (graph-capture-safe init).
// ---------------------------------------------------------------------------
__global__ void yolo_zero_mask(unsigned int* __restrict__ m, int n) {
    int i = blockIdx.x * blockDim.x + threadIdx.x;
    if (i < n) m[i] = 0u;
}

// ---------------------------------------------------------------------------
// Kernel 1: one thread per target. CIoU box loss + BCE class loss, plus
// setting the obj mask bit (idempotent atomicOr -> deterministic).
// ---------------------------------------------------------------------------
__global__ void yolo_targets(const float* __restrict__ pred,
                             const float* __restrict__ boxes,
                             const int* __restrict__ classes,
                             const float* __restrict__ anchors,
                             const int* __restrict__ stride_p,
                             float* __restrict__ out_box,
                             float* __restrict__ out_cls,
                             unsigned int* __restrict__ mask) {
    int i = blockIdx.x * blockDim.x + threadIdx.x;
    if (i >= NTGT) return;
    int b = i / NT;
    float s = (float)(*stride_p);

    const float* bx = boxes + (size_t)i * 4;
    float tx1 = bx[0], ty1 = bx[1], tx2 = bx[2], ty2 = bx[3];
    float cx = (tx1 + tx2) * 0.5f, cy = (ty1 + ty2) * 0.5f;
    float tw = tx2 - tx1, th = ty2 - ty1;
    float gx = cx / s, gy = cy / s, gw = tw / s, gh = th / s;
    int gi = (int)gx, gj = (int)gy;  // truncation, matches astype(int32) for x>=0
    bool valid = (gi >= 0) && (gi < NW) && (gj >= 0) && (gj < NH);
    int gic = min(max(gi, 0), NW - 1);
    int gjc = min(max(gj, 0), NH - 1);

    // anchor matching: argmin (first-min tie rule) of max(r, 1/r) over dims
    int best = 0;
    float bestp = 3.4e38f, aw = 1.0f, ah = 1.0f;
    for (int a = 0; a < NA; ++a) {
        float ax = anchors[a * 2 + 0], ay = anchors[a * 2 + 1];
        float rx = gw / ax, ry = gh / ay;
        float p = fmaxf(fmaxf(rx, 1.0f / rx), fmaxf(ry, 1.0f / ry));
        if (p < bestp) { bestp = p; best = a; aw = ax; ah = ay; }
    }

    if (!valid) { out_box[i] = 0.0f; out_cls[i] = 0.0f; return; }

    const float* pp = pred + ((size_t)(b * CHTOT + best * NCH)) * PLANE
                           + (size_t)gjc * NW + gic;
    float p0 = pp[0 * PLANE], p1 = pp[1 * PLANE];
    float p2 = pp[2 * PLANE], p3 = pp[3 * PLANE];

    float px = sigmoidf_(p0) + (float)gic;
    float py = sigmoidf_(p1) + (float)gjc;
    float pw = expf(p2) * aw;
    float ph = expf(p3) * ah;
    float px1 = (px - pw * 0.5f) * s, py1 = (py - ph * 0.5f) * s;
    float px2 = (px + pw * 0.5f) * s, py2 = (py + ph * 0.5f) * s;

    // CIoU
    float iw = fmaxf(fminf(px2, tx2) - fmaxf(px1, tx1), 0.0f);
    float ih = fmaxf(fminf(py2, ty2) - fmaxf(py1, ty1), 0.0f);
    float inter = iw * ih;
    float pa = (px2 - px1) * (py2 - py1);
    float ta = tw * th;
    float uni = pa + ta - inter + EPSF;
    float iou = inter / uni;
    float cw = fmaxf(px2, tx2) - fminf(px1, tx1);
    float chh = fmaxf(py2, ty2) - fminf(py1, ty1);
    float c2 = cw * cw + chh * chh + EPSF;
    float dx = px1 + px2 - tx1 - tx2;
    float dy = py1 + py2 - ty1 - ty2;
    float rho2 = (dx * dx + dy * dy) * 0.25f;
    float dat = atanf(tw / (th + EPSF)) - atanf((px2 - px1) / ((py2 - py1) + EPSF));
    float v = INV_PI2_4 * dat * dat;
    float alpha = v / (v - iou + 1.0f + EPSF);
    out_box[i] = 1.0f - iou + rho2 / c2 + alpha * v;

    // class BCE-with-logits vs one-hot
    int cls = classes[i];
    float lc = 0.0f;
    for (int c = 0; c < NC; ++c) {
        float logit = pp[(5 + c) * PLANE];
        float t = (c == cls) ? 1.0f : 0.0f;
        lc += softplusf(logit) - logit * t;
    }
    out_cls[i] = lc;

    int mbit = (b * NA + best) * PLANE + gjc * NW + gic;
    atomicOr(&mask[mbit >> 5], 1u << (mbit & 31));
}

// ---------------------------------------------------------------------------
// Kernel 2: objectness pass. One block per (b, a) plane (6400 contiguous f32).
// softplus sum is accumulated on the matrix pipe: V_WMMA_F32_16X16X4_F32 with
// B = ones gives D[m][n] = sum_k A[m][k] + C[m][n]; the grand total appears
// 16x in the accumulator, so dividing by 16 (exact) recovers it in full f32.
// All 32 lanes of a wave run a uniform chunk loop -> EXEC all-ones as WMMA
// requires. Reductions are fixed-order (shuffle + LDS) for determinism.
// ---------------------------------------------------------------------------
__global__ void yolo_obj(const float* __restrict__ pred,
                         const unsigned int* __restrict__ mask,
                         float* __restrict__ psp,
                         float* __restrict__ pmw) {
    int plane = blockIdx.x;            // 0..191
    int b = plane / NA, a = plane % NA;
    size_t base = ((size_t)(b * CHTOT + a * NCH + 4)) * PLANE;  // obj channel
    float wb = (float)(BATCH - b);
    int wave = threadIdx.x >> 5;
    int lane = threadIdx.x & 31;

    v8f acc = {0.f, 0.f, 0.f, 0.f, 0.f, 0.f, 0.f, 0.f};
    float sp_scalar = 0.0f;            // fallback path only
    float mw = 0.0f;

    const int nchunks = PLANE / 64;    // 100 chunks of 64 elements
    for (int c = wave; c < nchunks; c += 8) {
        int e0 = c * 64 + lane;
        int e1 = e0 + 32;
        float x0 = pred[base + e0];
        float x1 = pred[base + e1];
        float s0 = softplusf(x0);
        float s1 = softplusf(x1);
#if __has_builtin(__builtin_amdgcn_wmma_f32_16x16x4_f32)
        v2f av = {s0, s1};
        v2f bv = {1.0f, 1.0f};
        acc = __builtin_amdgcn_wmma_f32_16x16x4_f32(
            /*neg_a=*/false, av, /*neg_b=*/false, bv,
            /*c_mod=*/(short)0, acc, /*reuse_a=*/false, /*reuse_b=*/false);
#else
        sp_scalar += s0 + s1;
#endif
        int m0 = plane * PLANE + e0;
        if ((mask[m0 >> 5] >> (m0 & 31)) & 1u) mw += wb * x0;
        int m1 = m0 + 32;
        if ((mask[m1 >> 5] >> (m1 & 31)) & 1u) mw += wb * x1;
    }

    float s = acc[0] + acc[1] + acc[2] + acc[3] + acc[4] + acc[5] + acc[6] + acc[7];
    s = s * (1.0f / 16.0f) + sp_scalar;  // each row-sum replicated over 16 cols

    // fixed-order wave32 reduction
    for (int off = 16; off > 0; off >>= 1) {
        s  += __shfl_xor(s,  off, 32);
        mw += __shfl_xor(mw, off, 32);
    }
    __shared__ float ssp[8], smw[8];
    if (lane == 0) { ssp[wave] = s; smw[wave] = mw; }
    __syncthreads();
    if (threadIdx.x == 0) {
        float ts = 0.0f, tm = 0.0f;
        for (int w = 0; w < 8; ++w) { ts += ssp[w]; tm += smw[w]; }
        psp[plane] = ts;
        pmw[plane] = tm;
    }
}

// ---------------------------------------------------------------------------
// Kernel 3: deterministic final reduction and loss assembly.
// ---------------------------------------------------------------------------
__global__ void yolo_finalize(const float* __restrict__ box_l,
                              const float* __restrict__ cls_l,
                              const float* __restrict__ psp,
                              const float* __restrict__ pmw,
                              float* __restrict__ out) {
    __shared__ float s0[256], s1[256], s2[256], s3[256];
    int t = threadIdx.x;
    float a0 = 0.f, a1 = 0.f, a2 = 0.f, a3 = 0.f;
    for (int i = t; i < NTGT; i += 256) { a0 += box_l[i]; a1 += cls_l[i]; }
    for (int i = t; i < NPLANES; i += 256) { a2 += psp[i]; a3 += pmw[i]; }
    s0[t] = a0; s1[t] = a1; s2[t] = a2; s3[t] = a3;
    __syncthreads();
    for (int off = 128; off > 0; off >>= 1) {
        if (t < off) {
            s0[t] += s0[t + off];
            s1[t] += s1[t + off];
            s2[t] += s2[t + off];
            s3[t] += s3[t + off];
        }
        __syncthreads();
    }
    if (t == 0) {
        float loss_box = s0[0];
        float loss_cls = s1[0];
        // loss_obj = B*mean(softplus) - sum_b (B-b)*masked_sum_b / n_tot
        float loss_obj = ((float)BATCH * s2[0] - s3[0]) / (float)NTOT;
        float total = 5.0f * loss_box / (float)NTGT
                    + loss_obj / (float)BATCH
                    + 1.0f * loss_cls / (float)NTGT;
        out[0] = total;
    }
}

// ---------------------------------------------------------------------------
extern "C" void kernel_launch(void* const* d_in, const int* in_sizes, int n_in,
                              void* d_out, int out_size, void* d_ws, size_t ws_size,
                              hipStream_t stream) {
    const float* pred     = (const float*)d_in[0];
    const float* boxes    = (const float*)d_in[1];
    const int*   classes  = (const int*)d_in[2];
    const float* anchors  = (const float*)d_in[3];
    const int*   stride_p = (const int*)d_in[4];
    (void)in_sizes; (void)n_in; (void)out_size;

    // workspace layout (floats): box[3200] | cls[3200] | psp[192] | pmw[192] | mask bits
    float* ws    = (float*)d_ws;
    float* box_l = ws;
    float* cls_l = ws + NTGT;
    float* psp   = ws + 2 * NTGT;
    float* pmw   = psp + NPLANES;
    unsigned int* mask = (unsigned int*)(pmw + NPLANES);
    const int maskwords = (NTOT + 31) / 32;  // 38400 words
    (void)ws_size;  // ~181 KB total, well under the provided scratch

    yolo_zero_mask<<<(maskwords + 255) / 256, 256, 0, stream>>>(mask, maskwords);
    yolo_targets<<<(NTGT + 255) / 256, 256, 0, stream>>>(
        pred, boxes, classes, anchors, stride_p, box_l, cls_l, mask);
    yolo_obj<<<NPLANES, 256, 0, stream>>>(pred, mask, psp, pmw);
    yolo_finalize<<<1, 256, 0, stream>>>(box_l, cls_l, psp, pmw, (float*)d_out);
}